// QuantumNAT_Enhanced_65481071406977
// MI455X (gfx1250) — compile-verified
//
#include <hip/hip_runtime.h>
#include <hip/hip_bf16.h>

#define N_QUBITS 4
#define N_LAYERS 3

typedef float v2f __attribute__((ext_vector_type(2)));
typedef float v8f __attribute__((ext_vector_type(8)));

// ---------------------------------------------------------------------------
// Stage 1: build the stacked real matrices A[q][j][k] = Re( conj(d_j) *
// (V^H Z_q V)[j,k] * d_k ),  d_k = (-i)^popcount(k), V = variational unitary.
// Output: Amat[64][16] floats (4 KB) in workspace.
// Lanes 0..15 each own one column of V in registers (constant indexing after
// full unroll), write it to LDS once; then all 256 threads compute the 1024
// A entries (16-term complex dots, loops kept rolled for code size).
// ---------------------------------------------------------------------------
__global__ void qnat_prep(const float* __restrict__ w, float* __restrict__ Amat) {
    __shared__ float Vr[16][16];
    __shared__ float Vi[16][16];
    const int tid = threadIdx.x;

    if (tid < 16) {
        const int col = tid;
        float vr[16], vi[16];
#pragma unroll
        for (int r = 0; r < 16; ++r) { vr[r] = (r == col) ? 1.f : 0.f; vi[r] = 0.f; }

#pragma unroll
        for (int layer = 0; layer < N_LAYERS; ++layer) {
#pragma unroll
            for (int q = 0; q < N_QUBITS; ++q) {
                const float a  = w[(layer * 4 + q) * 3 + 0];
                const float b  = w[(layer * 4 + q) * 3 + 1];
                const float cz = w[(layer * 4 + q) * 3 + 2];
                float sa, ca, sb, cb, sc, cc;
                __sincosf(a * 0.5f, &sa, &ca);
                __sincosf(b * 0.5f, &sb, &cb);
                __sincosf(cz * 0.5f, &sc, &cc);
                // M1 = Ry(b) @ Rx(a)
                const float m00r =  cb * ca, m00i =  sb * sa;
                const float m01r = -sb * ca, m01i = -cb * sa;
                const float m10r =  sb * ca, m10i = -cb * sa;
                const float m11r =  cb * ca, m11i = -sb * sa;
                // U = Rz(cz) @ M1 : row0 *= (cc - i sc), row1 *= (cc + i sc)
                const float u00r = m00r * cc + m00i * sc, u00i = m00i * cc - m00r * sc;
                const float u01r = m01r * cc + m01i * sc, u01i = m01i * cc - m01r * sc;
                const float u10r = m10r * cc - m10i * sc, u10i = m10i * cc + m10r * sc;
                const float u11r = m11r * cc - m11i * sc, u11i = m11i * cc + m11r * sc;

                const int bit = 1 << (3 - q);   // qubit q <-> bit (3-q), MSB-first
#pragma unroll
                for (int k0 = 0; k0 < 16; ++k0) {
                    if (k0 & bit) continue;
                    const int k1 = k0 | bit;
                    const float ar = vr[k0], ai = vi[k0];
                    const float br = vr[k1], bi = vi[k1];
                    vr[k0] = u00r * ar - u00i * ai + u01r * br - u01i * bi;
                    vi[k0] = u00r * ai + u00i * ar + u01r * bi + u01i * br;
                    vr[k1] = u10r * ar - u10i * ai + u11r * br - u11i * bi;
                    vi[k1] = u10r * ai + u10i * ar + u11r * bi + u11i * br;
                }
            }
            // CNOT chain: control c -> target c+1 ; new V[r] = V[r ^ tbit] where cbit set
#pragma unroll
            for (int c = 0; c < N_QUBITS - 1; ++c) {
                const int cbit = 1 << (3 - c);
                const int tbit = 1 << (3 - (c + 1));
#pragma unroll
                for (int k = 0; k < 16; ++k) {
                    if ((k & cbit) && !(k & tbit)) {
                        const int k2 = k | tbit;
                        float t;
                        t = vr[k]; vr[k] = vr[k2]; vr[k2] = t;
                        t = vi[k]; vi[k] = vi[k2]; vi[k2] = t;
                    }
                }
            }
        }
#pragma unroll
        for (int r = 0; r < 16; ++r) { Vr[r][col] = vr[r]; Vi[r][col] = vi[r]; }
    }
    __syncthreads();

#pragma unroll 1
    for (int idx = tid; idx < 4 * 16 * 16; idx += blockDim.x) {
        const int q = idx >> 8;
        const int j = (idx >> 4) & 15;
        const int k = idx & 15;
        float mr = 0.f, mi = 0.f;
#pragma unroll 1
        for (int r = 0; r < 16; ++r) {
            const float z = ((r >> (3 - q)) & 1) ? -1.f : 1.f;
            // conj(V[r][j]) * V[r][k]
            mr += z * (Vr[r][j] * Vr[r][k] + Vi[r][j] * Vi[r][k]);
            mi += z * (Vr[r][j] * Vi[r][k] - Vi[r][j] * Vr[r][k]);
        }
        // multiply by (-i)^((pc(k)-pc(j)) mod 4), take real part
        const int d = (__popc(k) - __popc(j)) & 3;
        float re;
        if      (d == 0) re =  mr;
        else if (d == 1) re =  mi;
        else if (d == 2) re = -mr;
        else             re = -mi;
        Amat[(q * 16 + j) * 16 + k] = re;
    }
}

// ---------------------------------------------------------------------------
// Stage 2: one wave32 per 16-sample tile.
//   phi[k] = uv[k>>2] * wz[k&3]  (product-state amplitudes, factored form —
//   all per-lane fragment values derived with constant indexing + 4 selects)
//   T(64x16) = A(64x16) * Phi(16x16)  via 16x V_WMMA_F32_16X16X4_F32
//   out_q[s] = dot(T[16q..16q+15][s], phi[s])  (lane pair L, L^16 reduction)
// ---------------------------------------------------------------------------
__global__ void __launch_bounds__(256)
qnat_main(const float* __restrict__ x, const float* __restrict__ Amat,
          float* __restrict__ out, int B) {
    const int lane = threadIdx.x & 31;
    const int m    = lane & 15;            // matrix row / sample column within tile
    const bool hi  = (lane & 16) != 0;     // false: low half, true: high half
    const int wavesPerBlock = blockDim.x >> 5;
    const int waveId = blockIdx.x * wavesPerBlock + (threadIdx.x >> 5);
    const int totalWaves = gridDim.x * wavesPerBlock;
    const int nTiles = B >> 4;

    // A fragments: afrag[mt][kc] covers rows 16*mt..+15, cols 4*kc..+3.
    // 32-bit A 16x4 layout: lanes 0-15 hold K={0,1}, lanes 16-31 hold K={2,3}.
    v2f afrag[4][4];
#pragma unroll
    for (int mt = 0; mt < 4; ++mt) {
        const float* row = Amat + (16 * mt + m) * 16;
#pragma unroll
        for (int kc = 0; kc < 4; ++kc) {
            const int col = 4 * kc + (hi ? 2 : 0);
            afrag[mt][kc] = v2f{row[col], row[col + 1]};
        }
    }

    const float4* x4 = reinterpret_cast<const float4*>(x);

    for (int tile = waveId; tile < nTiles; tile += totalWaves) {
        const int s = tile * 16 + m;           // this lane's sample
        const float4 xv = x4[s];

        float c0, s0, c1, s1, c2, s2, c3, s3;
        __sincosf(xv.x * 0.5f, &s0, &c0);
        __sincosf(xv.y * 0.5f, &s1, &c1);
        __sincosf(xv.z * 0.5f, &s2, &c2);
        __sincosf(xv.w * 0.5f, &s3, &c3);

        // phi[k] = uv[k>>2] * wz[k&3],  k = b0*8 + b1*4 + b2*2 + b3 (qubit0 = MSB)
        const float uv0 = c0 * c1, uv1 = c0 * s1, uv2 = s0 * c1, uv3 = s0 * s1;
        const float wz0 = c2 * c3, wz1 = c2 * s3, wz2 = s2 * c3, wz3 = s2 * s3;
        const float uvq[4] = {uv0, uv1, uv2, uv3};
        const float wzq[4] = {wz0, wz1, wz2, wz3};

        // B fragments (4x16 f32): bfrag[kc] = { phi[4kc+2hi], phi[4kc+2hi+1] }
        //   = { uv[kc]*wz[2hi], uv[kc]*wz[2hi+1] }  -> only 2 selects on hi.
        const float wzA = hi ? wz2 : wz0;
        const float wzB = hi ? wz3 : wz1;
        v2f bfrag[4];
#pragma unroll
        for (int kc = 0; kc < 4; ++kc)
            bfrag[kc] = v2f{uvq[kc] * wzA, uvq[kc] * wzB};

        v8f acc[4];
#pragma unroll
        for (int mt = 0; mt < 4; ++mt)
            acc[mt] = v8f{0.f, 0.f, 0.f, 0.f, 0.f, 0.f, 0.f, 0.f};

#pragma unroll
        for (int mt = 0; mt < 4; ++mt) {
#pragma unroll
            for (int kc = 0; kc < 4; ++kc) {
                acc[mt] = __builtin_amdgcn_wmma_f32_16x16x4_f32(
                    /*neg_a=*/false, afrag[mt][kc],
                    /*neg_b=*/false, bfrag[kc],
                    /*c_mod=*/(short)0, acc[mt],
                    /*reuse_a=*/false, /*reuse_b=*/false);
            }
        }

        // D layout: lane L holds column N=L%16, VGPR v holds row M = v + 8*hi.
        // phi[v+8hi] = uv[(v>>2)+2hi] * wz[v&3]  -> 2 selects, constant indices.
        const float uvA = hi ? uv2 : uv0;
        const float uvB = hi ? uv3 : uv1;
        float phil[8];
#pragma unroll
        for (int v = 0; v < 8; ++v)
            phil[v] = ((v < 4) ? uvA : uvB) * wzq[v & 3];

        float res[4];
#pragma unroll
        for (int q = 0; q < 4; ++q) {
            float part = 0.f;
#pragma unroll
            for (int v = 0; v < 8; ++v) part += acc[q][v] * phil[v];
            part += __shfl_xor(part, 16, 32);
            res[q] = part;
        }
        if (lane < 16) {
            reinterpret_cast<float4*>(out)[s] = make_float4(res[0], res[1], res[2], res[3]);
        }
    }
}

extern "C" void kernel_launch(void* const* d_in, const int* in_sizes, int n_in,
                              void* d_out, int out_size, void* d_ws, size_t ws_size,
                              hipStream_t stream) {
    const float* x = (const float*)d_in[0];      // [B, 4]
    const float* w = (const float*)d_in[1];      // [3, 4, 3]
    float* out  = (float*)d_out;                 // [B, 4]
    float* Amat = (float*)d_ws;                  // 64*16 floats

    const int B = in_sizes[0] / N_QUBITS;

    qnat_prep<<<1, 256, 0, stream>>>(w, Amat);

    const int nTiles = B >> 4;
    int blocks = 1024;                           // 8192 waves, grid-stride over tiles
    const int maxBlocks = (nTiles + 7) / 8;
    if (blocks > maxBlocks) blocks = maxBlocks > 0 ? maxBlocks : 1;
    qnat_main<<<blocks, 256, 0, stream>>>(x, Amat, out, B);
}